// SequentialyDependentGater_75453985456507
// MI455X (gfx1250) — compile-verified
//
#include <hip/hip_runtime.h>
#include <hip/hip_bf16.h>
#include <math.h>

// Problem constants (match the reference)
#define BATCH 8
#define SEQ   8192
#define DIM   1024
#define FSZ   4
#define ROWS  (BATCH * SEQ)       // 65536 token rows
#define FWSTR 8                    // padded stride for fw rows (aligned)
#define CHUNK 128                  // steps per FSM chunk
#define NCHUNK (SEQ / CHUNK)       // 64 chunks per batch

typedef __attribute__((ext_vector_type(2))) float v2f;
typedef __attribute__((ext_vector_type(8))) float v8f;

// ---------------------------------------------------------------------------
// Kernel 1: fw[b,s,f] = x[b,s,:] . W[:,f]  via V_WMMA_F32_16X16X4_F32.
// Each wave computes one 16-row tile; W padded to 16 cols in LDS.
// A frag (16x4 f32): lanes 0-15 hold K=kk,kk+1 ; lanes 16-31 hold K=kk+2,kk+3.
// B frag (4x16 f32): mirrored layout; LDS stores (k,k+1) row pairs adjacent per
// column so a B fragment is a single aligned 8-byte LDS read.
// ---------------------------------------------------------------------------
__global__ __launch_bounds__(256) void gater_proj_wmma(
    const float* __restrict__ x,      // [ROWS, DIM]
    const float* __restrict__ W,      // [DIM, FSZ+1]
    float* __restrict__ fw)           // [ROWS, FWSTR] (cols 0..4 valid)
{
    // LDS: Wpad, element (k, n) at ((k>>1)*32 + 2*n + (k&1)); 1024*16 f32 = 64KB
    __shared__ float Wpad[DIM * 16];

    const int tid = threadIdx.x;
    for (int e = tid; e < DIM * 16; e += 256) {
        const int k = e >> 4;
        const int n = e & 15;
        const float v = (n < (FSZ + 1)) ? W[k * (FSZ + 1) + n] : 0.0f;
        Wpad[((k >> 1) * 32) + 2 * n + (k & 1)] = v;
    }
    __syncthreads();

    const int wave    = tid >> 5;                 // 8 waves per block
    const int lane    = tid & 31;
    const int m       = lane & 15;                // row within tile / B column
    const int hi      = lane >> 4;                // K-half select
    const int tile    = blockIdx.x * 8 + wave;    // 4096 tiles total
    const int rowBase = tile * 16;

    const float* xrow = x + (size_t)(rowBase + m) * DIM + 2 * hi;

    v8f c = {};
    for (int kk = 0; kk < DIM; kk += 4) {
        __builtin_prefetch((const void*)(xrow + kk + 128), 0, 1);
        // A fragment: two consecutive f32 of this lane's row at K = kk + 2*hi
        v2f a = *(const v2f*)(xrow + kk);
        // B fragment: Wpad rows (kk+2*hi, kk+2*hi+1) at column m
        v2f b = *(const v2f*)&Wpad[(((kk + 2 * hi) >> 1) * 32) + 2 * m];
        // 8 args: (neg_a, A, neg_b, B, c_mod, C, reuse_a, reuse_b)
        c = __builtin_amdgcn_wmma_f32_16x16x4_f32(
                false, a, false, b, (short)0, c, false, false);
    }

    // C layout: VGPR j -> (M = j + 8*hi, N = lane&15). Keep N = 0..4 only.
    if (m < (FSZ + 1)) {
        #pragma unroll
        for (int j = 0; j < 8; ++j) {
            const int row = rowBase + j + 8 * hi;
            fw[(size_t)row * FWSTR + m] = c[j];
        }
    }
}

// ---------------------------------------------------------------------------
// Shared step math (must be identical in passes A and C for bit-identical
// sampling decisions).
// ---------------------------------------------------------------------------
__device__ __forceinline__ float step_logit(const float* __restrict__ f,
                                            unsigned st)
{
    // state bits: bit0 = oldest window sample ... bit3 = newest
    return f[0]
         + f[1] * (float)( st       & 1u)
         + f[2] * (float)((st >> 1) & 1u)
         + f[3] * (float)((st >> 2) & 1u)
         + f[4] * (float)((st >> 3) & 1u);
}

__device__ __forceinline__ float step_tau(float u)
{
    // u < sigmoid(l)  <=>  log(u) - log1p(-u) < l
    return logf(u) - log1pf(-u);
}

// ---------------------------------------------------------------------------
// Kernel 2 (Pass A): per-(batch, chunk, entering-state) FSM replay -> 4-bit
// exit state. 8*64*16 = 8192 threads; the 16 state-threads of a group read
// identical fw/noise addresses (broadcast from L2).
// ---------------------------------------------------------------------------
__global__ __launch_bounds__(256) void gater_fsm_maps(
    const float* __restrict__ fw,         // [ROWS, FWSTR]
    const float* __restrict__ noise,      // [BATCH, SEQ]
    unsigned char* __restrict__ maps)     // [BATCH*NCHUNK, 16]
{
    const int gid = blockIdx.x * blockDim.x + threadIdx.x;
    if (gid >= BATCH * NCHUNK * 16) return;

    const int state = gid & 15;
    const int task  = gid >> 4;
    const int b     = task / NCHUNK;
    const int chunk = task % NCHUNK;
    const int t0    = chunk * CHUNK;

    unsigned cur = (unsigned)state;
    const float* fwb = fw    + (size_t)b * SEQ * FWSTR;
    const float* ub  = noise + (size_t)b * SEQ;

    for (int t = t0; t < t0 + CHUNK; ++t) {
        const float* f = fwb + (size_t)t * FWSTR;
        const float tau   = step_tau(ub[t]);
        const float logit = step_logit(f, cur);
        const unsigned samp = (logit > tau) ? 1u : 0u;
        cur = (cur >> 1) | (samp << 3);
    }
    maps[task * 16 + state] = (unsigned char)cur;
}

// ---------------------------------------------------------------------------
// Kernel 3 (Pass B): chase the known start state (all-zero window -> state 0)
// through the chunk maps. One lane per batch.
// ---------------------------------------------------------------------------
__global__ void gater_fsm_chain(
    const unsigned char* __restrict__ maps,   // [BATCH*NCHUNK, 16]
    unsigned char* __restrict__ enter)        // [BATCH, NCHUNK]
{
    const int b = threadIdx.x;
    if (b >= BATCH) return;
    unsigned st = 0;
    for (int c = 0; c < NCHUNK; ++c) {
        enter[b * NCHUNK + c] = (unsigned char)st;
        st = maps[(b * NCHUNK + c) * 16 + st];
    }
}

// ---------------------------------------------------------------------------
// Kernel 4 (Pass C): replay each chunk with its resolved entering state and
// emit (log_sigmoid, sigmoid, sample). 512 parallel serial chunks.
// ---------------------------------------------------------------------------
__global__ __launch_bounds__(256) void gater_fsm_emit(
    const float* __restrict__ fw,
    const float* __restrict__ noise,
    const unsigned char* __restrict__ enter,
    float* __restrict__ out)                  // [3, BATCH, SEQ]
{
    const int gid = blockIdx.x * blockDim.x + threadIdx.x;
    if (gid >= BATCH * NCHUNK) return;

    const int b     = gid / NCHUNK;
    const int chunk = gid % NCHUNK;
    const int t0    = chunk * CHUNK;

    unsigned st = enter[gid];
    const float* fwb = fw    + (size_t)b * SEQ * FWSTR;
    const float* ub  = noise + (size_t)b * SEQ;

    float* outLogp = out;
    float* outP    = out + (size_t)BATCH * SEQ;
    float* outSamp = out + (size_t)2 * BATCH * SEQ;

    for (int t = t0; t < t0 + CHUNK; ++t) {
        const float* f = fwb + (size_t)t * FWSTR;
        const float tau   = step_tau(ub[t]);
        const float logit = step_logit(f, st);

        // stable log_sigmoid / sigmoid
        const float logp = (logit >= 0.0f) ? -log1pf(expf(-logit))
                                           : (logit - log1pf(expf(logit)));
        const float p    = 1.0f / (1.0f + expf(-logit));
        const unsigned samp = (logit > tau) ? 1u : 0u;

        const size_t o = (size_t)b * SEQ + t;
        outLogp[o] = logp;
        outP[o]    = p;
        outSamp[o] = (float)samp;

        st = (st >> 1) | (samp << 3);
    }
}

// ---------------------------------------------------------------------------
extern "C" void kernel_launch(void* const* d_in, const int* in_sizes, int n_in,
                              void* d_out, int out_size, void* d_ws, size_t ws_size,
                              hipStream_t stream) {
    (void)in_sizes; (void)n_in; (void)out_size; (void)ws_size;

    const float* x     = (const float*)d_in[0];   // [B,S,D] f32
    const float* W     = (const float*)d_in[1];   // [D, 5]  f32
    const float* noise = (const float*)d_in[2];   // [B,S,1] f32
    float* out = (float*)d_out;                   // 3 * B * S f32

    // Workspace carve-up
    unsigned char* ws = (unsigned char*)d_ws;
    float*         fw    = (float*)ws;                                   // 2 MB
    unsigned char* maps  = ws + (size_t)ROWS * FWSTR * sizeof(float);    // 8 KB
    unsigned char* enter = maps + (size_t)BATCH * NCHUNK * 16;           // 512 B

    // 1) WMMA projection: 4096 tiles of 16 rows, 8 waves/block
    gater_proj_wmma<<<ROWS / 16 / 8, 256, 0, stream>>>(x, W, fw);

    // 2) FSM transition maps: 8*64*16 threads
    gater_fsm_maps<<<(BATCH * NCHUNK * 16 + 255) / 256, 256, 0, stream>>>(
        fw, noise, maps);

    // 3) Chase start state through maps
    gater_fsm_chain<<<1, 32, 0, stream>>>(maps, enter);

    // 4) Replay + emit outputs
    gater_fsm_emit<<<(BATCH * NCHUNK + 255) / 256, 256, 0, stream>>>(
        fw, noise, enter, out);
}